// SpectralConv2d_82686710382848
// MI455X (gfx1250) — compile-verified
//
#include <hip/hip_runtime.h>
#include <cstddef>
#include <cstdint>

// ---------------------------------------------------------------------------
// Spectral conv (FNO layer) for MI455X / gfx1250.
// All heavy math = pruned DFT-GEMMs on V_WMMA_F32_16X16X4_F32 (wave32).
// DFT / weight tables are staged into LDS with the Tensor Data Mover
// (tensor_load_to_lds + s_wait_tensorcnt), B operands are pair-interleaved
// so every WMMA B fetch is one 8-byte ds load.
// ---------------------------------------------------------------------------

typedef float    v2f __attribute__((ext_vector_type(2)));
typedef float    v8f __attribute__((ext_vector_type(8)));
typedef uint32_t v4u __attribute__((ext_vector_type(4)));
typedef uint32_t v8u __attribute__((ext_vector_type(8)));

__device__ __forceinline__ v8f wmma4(v2f a, v2f b, v8f c) {
  // D = A(16x4 f32) * B(4x16 f32) + C(16x16 f32)
  return __builtin_amdgcn_wmma_f32_16x16x4_f32(
      false, a, false, b, (short)0, c, false, false);
}

// ---------------------------------------------------------------------------
// TDM: DMA a contiguous run of n_floats f32 from global into LDS offset 0.
// D# group0: count=1 | lds_addr | global_addr[56:0] | type=2 ("image")
// D# group1: data_size=4B, 1-D tile (tile_dim0 = n, tile_dim1 unused)
// Issued by one wave; completion via TENSORcnt.
// ---------------------------------------------------------------------------
__device__ __forceinline__ void tdm_load_1d(uint32_t lds_byte_off,
                                            const void* gsrc,
                                            uint32_t n_floats) {
  const uint64_t ga = (uint64_t)(uintptr_t)gsrc;
  v4u g0;
  g0.x = 1u;                                       // count = 1 (valid D#)
  g0.y = lds_byte_off;                             // lds_addr
  g0.z = (uint32_t)ga;                             // global_addr[31:0]
  g0.w = ((uint32_t)(ga >> 32) & 0x01FFFFFFu)      // global_addr[56:32]
         | (2u << 30);                             // type = 2
  v8u g1;
  g1.s0 = 2u << 16;                                // data_size = 4 bytes
  g1.s1 = (n_floats & 0xFFFFu) << 16;              // tensor_dim0[15:0]
  g1.s2 = (n_floats >> 16) & 0xFFFFu;              // tensor_dim0[31:16], dim1=0
  g1.s3 = (n_floats & 0xFFFFu) << 16;              // tile_dim0 = n (<= 65535)
  g1.s4 = 0u;                                      // tile_dim1/2 unused
  g1.s5 = n_floats;                                // tensor_dim0_stride lo
  g1.s6 = 0u;
  g1.s7 = 0u;
  asm volatile("tensor_load_to_lds %0, %1" :: "s"(g0), "s"(g1) : "memory");
}

__device__ __forceinline__ void stage_table_to_lds(const float* gsrc,
                                                   uint32_t n_floats) {
  if (threadIdx.x < 32u) {                         // wave 0 issues the DMA
    tdm_load_1d(0u, gsrc, n_floats);
    __builtin_amdgcn_s_wait_tensorcnt(0);
  }
  __syncthreads();
}

// Workspace layout (float offsets). Tables are PAIR-INTERLEAVED:
//   P[k>>1][n][2] holds B[k][n] (slot k&1) so one v2f load = B[kk],B[kk+1].
#define OFF_BW 0u                  //  128p x 32n x 2  fwd-W
#define OFF_BH 8192u               //  256p x 64n x 2  fwd-H
#define OFF_BI 40960u              //   32p x 512n x 2 inv-H
#define OFF_BV 73728u              //   16p x 256n x 2 inv-W (real)
#define OFF_WP 81920u              //  512 modes x 2 planes x 64c x 64d x 2
#define OFF_XW 8470528u            //  262144 x 32   (b,c,h) x (q | q_im)
#define OFF_XF 16859136u           //  16384  x 64   (b,c,q) x (t | t_im)
#define OFF_YF 17907712u           //  2 planes x 512 modes x 16b x 64d
#define OFF_YH 18956288u           //  262144 x 32   (b,d,h) x (q | q_im)
#define YF_PLANE 524288u

// ---------------------------------------------------------------------------
// Twiddle tables (ortho: 1/16 folded per 1-D stage), pair-interleaved.
// ---------------------------------------------------------------------------
__global__ void twiddle_init(float* __restrict__ ws) {
  const int idx = blockIdx.x * blockDim.x + threadIdx.x;
  const float STEP = 6.28318530717958647692f / 256.0f;
  const float S = 1.0f / 16.0f;

  if (idx < 8192) {                       // Bw[k=0..255][n=0..31]
    int k = idx >> 5, n = idx & 31, q = n & 15, ri = n >> 4;
    float s, c; __sincosf(STEP * (float)((k * q) & 255), &s, &c);
    ws[OFF_BW + (((k >> 1) * 32 + n) << 1) + (k & 1)] = (ri ? -s : c) * S;
  }
  if (idx < 32768) {                      // Bh[k=0..511][n=0..63]
    int k = idx >> 6, n = idx & 63;
    int h = k >> 1, kri = k & 1;
    int ti = n & 31, ro = n >> 5;
    int t = (ti < 16) ? ti : (224 + ti);
    float s, c; __sincosf(STEP * (float)((h * t) & 255), &s, &c);
    float er = c * S, ei = -s * S;        // e^{-i theta}
    ws[OFF_BH + (((k >> 1) * 64 + n) << 1) + (k & 1)] =
        (ro == 0) ? (kri ? -ei : er) : (kri ? er : ei);
  }
  if (idx < 32768) {                      // Bi[k=0..63][n=0..511]
    int k = idx >> 9, n = idx & 511;
    int ti = k >> 1, kri = k & 1;
    int h = n & 255, ro = n >> 8;
    int t = (ti < 16) ? ti : (224 + ti);
    float s, c; __sincosf(STEP * (float)((t * h) & 255), &s, &c);
    float er = c * S, ei = s * S;         // e^{+i theta}
    ws[OFF_BI + (((k >> 1) * 512 + n) << 1) + (k & 1)] =
        (ro == 0) ? (kri ? -ei : er) : (kri ? er : ei);
  }
  if (idx < 8192) {                       // Bv[k=0..31][n=0..255]
    int k = idx >> 8, n = idx & 255, q = k & 15, ri = k >> 4;
    float s, c; __sincosf(STEP * (float)((q * n) & 255), &s, &c);
    float amp = ((q == 0) ? 1.0f : 2.0f) * S;
    ws[OFF_BV + (((k >> 1) * 256 + n) << 1) + (k & 1)] =
        ri ? (-amp * s) : (amp * c);
  }
}

// ---------------------------------------------------------------------------
// Repack weights into GEMM-ready, sign-folded B' matrices:
//   WP[mode][plane][c][d][2]:  plane0 (re): (wr, -wi)   plane1 (im): (wi, wr)
// 64 KB per mode, contiguous -> one TDM load in stage 3.
// ---------------------------------------------------------------------------
__global__ void repack_w(const float* __restrict__ W1,
                         const float* __restrict__ W2,
                         float* __restrict__ WP) {
  const int idx = blockIdx.x * blockDim.x + threadIdx.x;  // 512*64*64
  const int d = idx & 63, c = (idx >> 6) & 63, mode = idx >> 12;
  const int t_idx = mode >> 4, q = mode & 15, tw = t_idx & 15;
  const float* __restrict__ Wt = (t_idx < 16) ? W1 : W2;
  const float* wp = Wt + ((size_t)((d * 64 + c) * 16 + tw) * 16u + q) * 2u;
  const float wr = wp[0], wi = wp[1];
  const size_t b0 = ((size_t)(mode * 2 + 0) * 64u + c) * 128u + d * 2u;
  const size_t b1 = ((size_t)(mode * 2 + 1) * 64u + c) * 128u + d * 2u;
  WP[b0] = wr;  WP[b0 + 1] = -wi;
  WP[b1] = wi;  WP[b1 + 1] = wr;
}

// ---------------------------------------------------------------------------
// Stage 1: Xw[(b,c,h)][q | q+16] = sum_w x * Bw.  M=262144 K=256 N=32.
// Bw (32 KB) staged in LDS via TDM.
// ---------------------------------------------------------------------------
__global__ void stage1_dftW(const float* __restrict__ x,
                            const float* __restrict__ BwG,
                            float* __restrict__ Xw) {
  extern __shared__ float smem[];                    // 8192 floats
  stage_table_to_lds(BwG, 8192u);

  const int wid  = blockIdx.x * 8 + (threadIdx.x >> 5);
  const int lane = threadIdx.x & 31;
  const int l15 = lane & 15, half = lane >> 4;

  const float* __restrict__ arow = x + (size_t)(wid * 16 + l15) * 256u;
  const v2f* __restrict__ bp = (const v2f*)smem;
  v8f acc0 = {}; v8f acc1 = {};

  for (int k = 0; k < 256; k += 4) {
    if (k + 64 < 256) __builtin_prefetch(arow + k + 64, 0, 3);
    const int p = (k >> 1) + half;                   // pair index
    v2f a = *(const v2f*)(arow + k + 2 * half);
    acc0 = wmma4(a, bp[p * 32 + l15],      acc0);
    acc1 = wmma4(a, bp[p * 32 + 16 + l15], acc1);
  }
  float* __restrict__ orow = Xw + (size_t)wid * 512u;
  #pragma unroll
  for (int j = 0; j < 8; ++j) {
    int m = j + 8 * half;
    orow[(size_t)m * 32u + l15]      = acc0[j];
    orow[(size_t)m * 32u + 16 + l15] = acc1[j];
  }
}

// ---------------------------------------------------------------------------
// Stage 2: Xf[(b,c,q)][t | t+32] = sum_h Xw * Bh (complex, K=512 interleaved).
// Bh (128 KB) staged in LDS via TDM; one wave = one (b,c).
// ---------------------------------------------------------------------------
__global__ void stage2_dftH(const float* __restrict__ Xw,
                            const float* __restrict__ BhG,
                            float* __restrict__ Xf) {
  extern __shared__ float smem[];                    // 32768 floats
  stage_table_to_lds(BhG, 32768u);

  const int bc   = blockIdx.x * 8 + (threadIdx.x >> 5);
  const int lane = threadIdx.x & 31;
  const int l15 = lane & 15, half = lane >> 4;

  const float* __restrict__ xb = Xw + (size_t)bc * 8192u;
  const v2f* __restrict__ bp = (const v2f*)smem;
  v8f acc[4] = {};

  for (int k = 0; k < 512; k += 4) {
    const int p = (k >> 1) + half;                   // pair index, h = p
    v2f a; a.x = xb[(size_t)p * 32u + l15];          // re
           a.y = xb[(size_t)p * 32u + 16u + l15];    // im
    #pragma unroll
    for (int nt = 0; nt < 4; ++nt)
      acc[nt] = wmma4(a, bp[p * 64 + nt * 16 + l15], acc[nt]);
  }
  float* __restrict__ orow = Xf + (size_t)bc * 1024u;
  #pragma unroll
  for (int nt = 0; nt < 4; ++nt)
    #pragma unroll
    for (int j = 0; j < 8; ++j) {
      int m = j + 8 * half;
      orow[(size_t)m * 64u + nt * 16 + l15] = acc[nt][j];
    }
}

// ---------------------------------------------------------------------------
// Stage 3: mode mixing. One BLOCK = one mode; its 64 KB B' tile is TDM-loaded
// into LDS; 8 waves split {plane(re/im)} x {16-wide d tile}.
//   out[b,d] = sum_c x[b,c] * w[d,c]   (complex, K interleaved = 128)
// ---------------------------------------------------------------------------
__global__ void stage3_mix(const float* __restrict__ Xf,
                           const float* __restrict__ WP,
                           float* __restrict__ Yf) {
  extern __shared__ float smem[];                    // 16384 floats (64 KB)
  const int mode = blockIdx.x;
  stage_table_to_lds(WP + (size_t)mode * 16384u, 16384u);

  const int wave = threadIdx.x >> 5;
  const int lane = threadIdx.x & 31;
  const int l15 = lane & 15, half = lane >> 4;
  const int plane = wave >> 2, nt = wave & 3;
  const int t_idx = mode >> 4, q = mode & 15;

  // Preload A: rows b = l15, K = 128 (2c+ri interleaved), 32 k-steps.
  v2f areg[32];
  #pragma unroll
  for (int s = 0; s < 32; ++s) {
    const int c = 2 * s + half;
    const size_t ib = ((size_t)(l15 * 64 + c) * 16u + q) * 64u + t_idx;
    areg[s].x = Xf[ib];                              // re
    areg[s].y = Xf[ib + 32u];                        // im
  }

  const v2f* __restrict__ bp =
      (const v2f*)smem + (size_t)plane * 4096u + nt * 16 + l15;
  v8f acc = {};
  for (int s = 0; s < 32; ++s) {
    const int c = 2 * s + half;
    acc = wmma4(areg[s], bp[c * 64], acc);
  }
  float* __restrict__ op = Yf + (size_t)plane * YF_PLANE
                              + (size_t)mode * 1024u;
  #pragma unroll
  for (int j = 0; j < 8; ++j) {
    int m = j + 8 * half;                            // m = b
    op[(size_t)m * 64u + nt * 16 + l15] = acc[j];
  }
}

// ---------------------------------------------------------------------------
// Stage 4: inverse H from 32 modes. K=64, N=512 (8 col groups).
// Bi (128 KB) staged in LDS via TDM.
// ---------------------------------------------------------------------------
__global__ void stage4_idftH(const float* __restrict__ Yf,
                             const float* __restrict__ BiG,
                             float* __restrict__ Yh) {
  extern __shared__ float smem[];                    // 32768 floats
  stage_table_to_lds(BiG, 32768u);

  const int gw   = blockIdx.x * 8 + (threadIdx.x >> 5);
  const int lane = threadIdx.x & 31;
  const int l15 = lane & 15, half = lane >> 4;

  const int bd = gw >> 3;                            // (b*64+d)
  const int g  = gw & 7;                             // column group of 64
  const int b  = bd >> 6, d = bd & 63;

  v2f areg[16];
  #pragma unroll
  for (int s = 0; s < 16; ++s) {
    const int ti = 2 * s + half;                     // t index pair
    const size_t ib = ((size_t)(ti * 16 + l15) * 16u + b) * 64u + d;
    areg[s].x = Yf[ib];                              // re plane
    areg[s].y = Yf[ib + YF_PLANE];                   // im plane
  }

  const v2f* __restrict__ bbase = (const v2f*)smem;
  v8f acc[4] = {};
  for (int s = 0; s < 16; ++s) {
    const int p = 2 * s + half;
    #pragma unroll
    for (int nt = 0; nt < 4; ++nt)
      acc[nt] = wmma4(areg[s], bbase[p * 512 + g * 64 + nt * 16 + l15], acc[nt]);
  }
  const int ro = g >> 2;
  const int hbase = (g & 3) * 64;
  #pragma unroll
  for (int nt = 0; nt < 4; ++nt)
    #pragma unroll
    for (int j = 0; j < 8; ++j) {
      int m = j + 8 * half;                          // m = q
      int h = hbase + nt * 16 + l15;
      Yh[((size_t)bd * 256u + h) * 32u + m + 16 * ro] = acc[nt][j];
    }
}

// ---------------------------------------------------------------------------
// Stage 5: inverse real W from 16 modes. K=32, N=256 (4 col groups).
// Bv (32 KB) staged in LDS via TDM.
// ---------------------------------------------------------------------------
__global__ void stage5_idftW(const float* __restrict__ Yh,
                             const float* __restrict__ Bv,
                             float* __restrict__ out) {
  extern __shared__ float smem[];                    // 8192 floats
  stage_table_to_lds(Bv, 8192u);

  const int gw   = blockIdx.x * 8 + (threadIdx.x >> 5);
  const int lane = threadIdx.x & 31;
  const int l15 = lane & 15, half = lane >> 4;

  const int tile = gw >> 2;
  const int g    = gw & 3;

  const float* __restrict__ ar = Yh + ((size_t)tile * 16u + l15) * 32u;
  v2f areg[8];
  #pragma unroll
  for (int s = 0; s < 8; ++s)
    areg[s] = *(const v2f*)(ar + 4 * s + 2 * half);

  const v2f* __restrict__ bbase = (const v2f*)smem;
  v8f acc[4] = {};
  #pragma unroll
  for (int s = 0; s < 8; ++s) {
    const int p = 2 * s + half;
    #pragma unroll
    for (int nt = 0; nt < 4; ++nt)
      acc[nt] = wmma4(areg[s], bbase[p * 256 + g * 64 + nt * 16 + l15], acc[nt]);
  }
  float* __restrict__ orow = out + (size_t)tile * 4096u;
  #pragma unroll
  for (int nt = 0; nt < 4; ++nt)
    #pragma unroll
    for (int j = 0; j < 8; ++j) {
      int m = j + 8 * half;
      orow[(size_t)m * 256u + g * 64 + nt * 16 + l15] = acc[nt][j];
    }
}

// ---------------------------------------------------------------------------
// Host launcher
// ---------------------------------------------------------------------------
extern "C" void kernel_launch(void* const* d_in, const int* in_sizes, int n_in,
                              void* d_out, int out_size, void* d_ws, size_t ws_size,
                              hipStream_t stream) {
  (void)in_sizes; (void)n_in; (void)out_size; (void)ws_size;
  const float* x  = (const float*)d_in[0];
  const float* w1 = (const float*)d_in[1];
  const float* w2 = (const float*)d_in[2];
  float* out = (float*)d_out;
  float* ws  = (float*)d_ws;

  float* Bw = ws + OFF_BW;
  float* Bh = ws + OFF_BH;
  float* Bi = ws + OFF_BI;
  float* Bv = ws + OFF_BV;
  float* WP = ws + OFF_WP;
  float* Xw = ws + OFF_XW;
  float* Xf = ws + OFF_XF;
  float* Yf = ws + OFF_YF;
  float* Yh = ws + OFF_YH;

  twiddle_init<<<128,  256, 0,      stream>>>(ws);
  repack_w    <<<8192, 256, 0,      stream>>>(w1, w2, WP);
  stage1_dftW <<<2048, 256, 32768,  stream>>>(x,  Bw, Xw);
  stage2_dftH <<<128,  256, 131072, stream>>>(Xw, Bh, Xf);
  stage3_mix  <<<512,  256, 65536,  stream>>>(Xf, WP, Yf);
  stage4_idftH<<<1024, 256, 131072, stream>>>(Yf, Bi, Yh);
  stage5_idftW<<<8192, 256, 32768,  stream>>>(Yh, Bv, out);
}